// GlobalGradMixtureEncoder_67551245631811
// MI455X (gfx1250) — compile-verified
//
#include <hip/hip_runtime.h>

typedef __attribute__((ext_vector_type(2))) float v2f;
typedef __attribute__((ext_vector_type(8))) float v8f;

#define Bn 256
#define Dn 512
#define Kn 512
#define En 256
#define PADROW (Kn + 4)

__constant__ float kBETA = 0.001f;

__device__ __forceinline__ v8f wmma4(v2f a, v2f b, v8f c) {
  // D = A(16x4,f32) * B(4x16,f32) + C(16x16,f32)
  return __builtin_amdgcn_wmma_f32_16x16x4_f32(false, a, false, b, (short)0, c,
                                               false, false);
}

__device__ __forceinline__ void row_softmax(float* row, int lane) {
  float mx = -3.0e38f;
  for (int j = lane; j < Kn; j += 32) mx = fmaxf(mx, row[j]);
  for (int off = 16; off; off >>= 1) mx = fmaxf(mx, __shfl_xor(mx, off, 32));
  float s = 0.f;
  for (int j = lane; j < Kn; j += 32) {
    float e = __expf(row[j] - mx);
    row[j] = e;
    s += e;
  }
  for (int off = 16; off; off >>= 1) s += __shfl_xor(s, off, 32);
  float inv = 1.f / s;
  for (int j = lane; j < Kn; j += 32) row[j] *= inv;
}

// colsq[k] = sum_d wimg[d,k]^2 ; rowsq[k] = sum_e wrec[k,e]^2
__global__ void k_sq(const float* __restrict__ wimg,
                     const float* __restrict__ wrec,
                     float* __restrict__ colsq, float* __restrict__ rowsq) {
  int k = blockIdx.x * blockDim.x + threadIdx.x;
  if (k < Kn) {
    float s = 0.f;
    for (int d = 0; d < Dn; ++d) { float w = wimg[d * Kn + k]; s += w * w; }
    colsq[k] = s;
    float r = 0.f;
    for (int e = 0; e < En; ++e) { float w = wrec[k * En + e]; r += w * w; }
    rowsq[k] = r;
  }
}

// G = wimg^T @ wimg  [K,K]; one 16x16 tile per wave
__global__ void k_gram(const float* __restrict__ wimg, float* __restrict__ G) {
  int wv   = blockIdx.x * (blockDim.x >> 5) + (threadIdx.x >> 5);
  int lane = threadIdx.x & 31;
  int l16 = lane & 15, half = lane >> 4;
  int i0 = (wv >> 5) * 16, j0 = (wv & 31) * 16;
  v8f c = {};
  for (int d0 = 0; d0 < Dn; d0 += 4) {
    int dd = d0 + 2 * half;
    v2f a, b;
    a.x = wimg[dd * Kn + i0 + l16];
    a.y = wimg[(dd + 1) * Kn + i0 + l16];
    b.x = wimg[dd * Kn + j0 + l16];
    b.y = wimg[(dd + 1) * Kn + j0 + l16];
    c = wmma4(a, b, c);
  }
  for (int v = 0; v < 8; ++v)
    G[(i0 + v + 8 * half) * Kn + j0 + l16] = c[v];
}

// Init: S = images@wimg (stored), xp = softmax((2S-colsq)*BETA/D), lat = xp@wrec
__global__ void k_init(const float* __restrict__ images,
                       const float* __restrict__ wimg,
                       const float* __restrict__ wrec,
                       const float* __restrict__ colsq,
                       float* __restrict__ S, float* __restrict__ lat) {
  __shared__ float xp_s[16][PADROW];
  int tid = threadIdx.x, wv = tid >> 5, lane = tid & 31;
  int l16 = lane & 15, half = lane >> 4;
  int r0 = blockIdx.x * 16;
  const float cD = kBETA / (float)Dn;

  for (int t = 0; t < 2; ++t) {                 // 16 waves * 2 tiles = K/16
    int n0 = (wv * 2 + t) * 16;
    v8f c = {};
    for (int d0 = 0; d0 < Dn; d0 += 4) {
      int dd = d0 + 2 * half;
      v2f a = *(const v2f*)&images[(r0 + l16) * Dn + dd];
      v2f b;
      b.x = wimg[dd * Kn + n0 + l16];
      b.y = wimg[(dd + 1) * Kn + n0 + l16];
      c = wmma4(a, b, c);
    }
    for (int v = 0; v < 8; ++v) {
      int m = v + 8 * half, n = n0 + l16;
      S[(r0 + m) * Kn + n] = c[v];
      xp_s[m][n] = (2.f * c[v] - colsq[n]) * cD;
    }
  }
  __syncthreads();
  row_softmax(&xp_s[wv][0], lane);              // wave wv owns row wv
  __syncthreads();

  int e0 = wv * 16;                             // lat = xp @ wrec, tile per wave
  v8f c = {};
  for (int k0 = 0; k0 < Kn; k0 += 4) {
    int kk = k0 + 2 * half;
    v2f a, b;
    a.x = xp_s[l16][kk];
    a.y = xp_s[l16][kk + 1];
    b.x = wrec[kk * En + e0 + l16];
    b.y = wrec[(kk + 1) * En + e0 + l16];
    c = wmma4(a, b, c);
  }
  for (int v = 0; v < 8; ++v)
    lat[(r0 + v + 8 * half) * En + e0 + l16] = c[v];
}

// One full refinement iteration, 16 rows per block.
__global__ void k_step(const float* __restrict__ S, const float* __restrict__ G,
                       const float* __restrict__ wrec,
                       const float* __restrict__ rowsq,
                       float* __restrict__ lat, float* __restrict__ LK) {
  __shared__ float xp_s[16][PADROW];            // holds xp, then g
  __shared__ float rsum[16];
  int tid = threadIdx.x, wv = tid >> 5, lane = tid & 31;
  int l16 = lane & 15, half = lane >> 4;
  int r0 = blockIdx.x * 16;
  const float cE = kBETA / (float)En;
  const float lr = kBETA * kBETA * 0.001f;

  // Phase 1: T = lat @ wrec^T  -> logits -> softmax -> xp (LDS)
  for (int t = 0; t < 2; ++t) {
    int n0 = (wv * 2 + t) * 16;
    v8f c = {};
    for (int e0 = 0; e0 < En; e0 += 4) {
      int ee = e0 + 2 * half;
      v2f a = *(const v2f*)&lat[(r0 + l16) * En + ee];
      v2f b = *(const v2f*)&wrec[(n0 + l16) * En + ee];
      c = wmma4(a, b, c);
    }
    for (int v = 0; v < 8; ++v) {
      int m = v + 8 * half, n = n0 + l16;
      xp_s[m][n] = (2.f * c[v] - rowsq[n]) * cE;
    }
  }
  __syncthreads();
  row_softmax(&xp_s[wv][0], lane);
  __syncthreads();

  // Phase 2: lk = S - xp @ G   (lk spilled to L2-resident scratch)
  for (int t = 0; t < 2; ++t) {
    int n0 = (wv * 2 + t) * 16;
    v8f c = {};
    for (int k0 = 0; k0 < Kn; k0 += 4) {
      int kk = k0 + 2 * half;
      v2f a, b;
      a.x = xp_s[l16][kk];
      a.y = xp_s[l16][kk + 1];
      b.x = G[kk * Kn + n0 + l16];
      b.y = G[(kk + 1) * Kn + n0 + l16];
      c = wmma4(a, b, c);
    }
    for (int v = 0; v < 8; ++v) {
      int m = v + 8 * half, n = n0 + l16;
      LK[(r0 + m) * Kn + n] = S[(r0 + m) * Kn + n] - c[v];
    }
  }
  __syncthreads();

  // Per-row: t1 = <lk,xp>; g = xp*(lk - t1) overwrites xp; rsum = sum_k g
  {
    int row = wv;
    const float* lkr = &LK[(r0 + row) * Kn];
    float t1 = 0.f;
    for (int j = lane; j < Kn; j += 32) t1 += lkr[j] * xp_s[row][j];
    for (int off = 16; off; off >>= 1) t1 += __shfl_xor(t1, off, 32);
    float sg = 0.f;
    for (int j = lane; j < Kn; j += 32) {
      float gv = xp_s[row][j] * (lkr[j] - t1);
      xp_s[row][j] = gv;
      sg += gv;
    }
    for (int off = 16; off; off >>= 1) sg += __shfl_xor(sg, off, 32);
    if (lane == 0) rsum[row] = sg;
  }
  __syncthreads();

  // Phase 3: latg = g @ wrec;  lat += LR*(latg - lat*rsum)
  int e0 = wv * 16;
  v8f c = {};
  for (int k0 = 0; k0 < Kn; k0 += 4) {
    int kk = k0 + 2 * half;
    v2f a, b;
    a.x = xp_s[l16][kk];
    a.y = xp_s[l16][kk + 1];
    b.x = wrec[kk * En + e0 + l16];
    b.y = wrec[(kk + 1) * En + e0 + l16];
    c = wmma4(a, b, c);
  }
  for (int v = 0; v < 8; ++v) {
    int m = v + 8 * half;
    int idx = (r0 + m) * En + e0 + l16;
    float lv = lat[idx];
    lat[idx] = lv + lr * (c[v] - lv * rsum[m]);
  }
}

extern "C" void kernel_launch(void* const* d_in, const int* in_sizes, int n_in,
                              void* d_out, int out_size, void* d_ws,
                              size_t ws_size, hipStream_t stream) {
  (void)in_sizes; (void)n_in; (void)out_size; (void)ws_size;
  const float* images = (const float*)d_in[0];
  const float* wimg   = (const float*)d_in[1];
  const float* wrec   = (const float*)d_in[2];

  float* ws    = (float*)d_ws;
  float* S     = ws;                    // B*K
  float* G     = S + Bn * Kn;           // K*K
  float* colsq = G + Kn * Kn;           // K
  float* rowsq = colsq + Kn;            // K
  float* lat   = rowsq + Kn;            // B*E
  float* LK    = lat + Bn * En;         // B*K   (~2.4 MB total, L2-resident)

  k_sq<<<2, 256, 0, stream>>>(wimg, wrec, colsq, rowsq);
  k_gram<<<128, 256, 0, stream>>>(wimg, G);
  k_init<<<Bn / 16, 512, 0, stream>>>(images, wimg, wrec, colsq, S, lat);
  for (int it = 0; it < 10; ++it)
    k_step<<<Bn / 16, 512, 0, stream>>>(S, G, wrec, rowsq, lat, LK);
  hipMemcpyAsync(d_out, lat, Bn * En * sizeof(float),
                 hipMemcpyDeviceToDevice, stream);
}